// ALIGNNEdge_2156073582923
// MI455X (gfx1250) — compile-verified
//
#include <hip/hip_runtime.h>
#include <cstddef>

typedef float v2f __attribute__((ext_vector_type(2)));
typedef float v4f __attribute__((ext_vector_type(4)));
typedef float v8f __attribute__((ext_vector_type(8)));

#define DEVINL __device__ __forceinline__

DEVINL float sigmoid_f(float v) { return 1.0f / (1.0f + expf(-v)); }
DEVINL float softplus_f(float v) { return fmaxf(v, 0.0f) + log1pf(expf(-fabsf(v))); }
DEVINL float silu_f(float v) { return v * sigmoid_f(v); }

// ---------------------------------------------------------------------------
// Tall-skinny GEMM: out[M,NCOL] = A[M,K] @ W[K,NCOL] + bias, via
// V_WMMA_F32_16X16X4_F32 (fp32 A/B, fp32 accumulate).
// Block = 256 threads (8 wave32), computes 64 rows x NCOL.
// B fragments (W) are preloaded once per wave into registers (they do not
// depend on the row-slab), so the steady-state loop is LDS-read + WMMA only.
// RBF=true: A is a 1-D array d[M]; features A[r,k]=exp(-gamma*(d-c0-k*dc)^2)
// are generated directly into the LDS A-tile (fuses the RBF expansion).
// ---------------------------------------------------------------------------
template <int K, int NCOL, bool RBF>
__global__ __launch_bounds__(256) void linear_wmma(
    const float* __restrict__ A, const float* __restrict__ W,
    const float* __restrict__ bias, float* __restrict__ out,
    int M, float c0, float dc, float gamma) {
  constexpr int KP = K + 4;               // padded LDS row stride (16B aligned)
  __shared__ float As[64 * KP];
  const int tid = threadIdx.x;
  const int rowBlock = blockIdx.x * 64;

  if (RBF) {
    for (int t = tid; t < 64 * K; t += 256) {
      int r = t / K, k = t - r * K;
      int gr = rowBlock + r;
      float v = 0.0f;
      if (gr < M) {
        float diff = A[gr] - (c0 + k * dc);
        v = expf(-gamma * diff * diff);
      }
      As[r * KP + k] = v;
    }
  } else {
    constexpr int KQ = K / 4;             // all K here are multiples of 4
    for (int t = tid; t < 64 * KQ; t += 256) {
      int r = t / KQ, kq = t - r * KQ;
      int gr = rowBlock + r;
      v4f vv = {};
      if (gr < M) vv = *(const v4f*)(A + (size_t)gr * K + kq * 4);
      *(v4f*)(&As[r * KP + kq * 4]) = vv;
    }
  }

  const int lane = tid & 31;
  const int wave = tid >> 5;
  constexpr int NTC = NCOL / 16;          // column tiles
  const int ct = wave % NTC;
  const int col = ct * 16 + (lane & 15);
  const int khalf = (lane >> 4) * 2;      // lanes 0-15 -> k+{0,1}; 16-31 -> k+{2,3}

  // Preload this wave's B fragments (shared across all row slabs).
  v2f bfr[K / 4];
#pragma unroll
  for (int kk = 0; kk < K; kk += 4) {
    bfr[kk / 4].x = W[(kk + khalf) * NCOL + col];       // B[k0,   col]
    bfr[kk / 4].y = W[(kk + khalf + 1) * NCOL + col];   // B[k0+1, col]
  }

  __syncthreads();

  for (int s = wave / NTC; s < 4; s += 8 / NTC) {
    int rbase = rowBlock + s * 16;
    if (rbase >= M) break;
    v8f acc = {};
    const float* arow = &As[(s * 16 + (lane & 15)) * KP];
#pragma unroll
    for (int kk = 0; kk < K; kk += 4) {
      v2f a = *(const v2f*)(arow + kk + khalf);         // A[row, k0..k0+1]
      acc = __builtin_amdgcn_wmma_f32_16x16x4_f32(
          false, a, false, bfr[kk / 4], (short)0, acc, false, false);
    }
    float bv = bias[col];
    int r0 = rbase + (lane >> 4) * 8;                   // C/D layout
#pragma unroll
    for (int v = 0; v < 8; v++) {
      int r = r0 + v;
      if (r < M) out[(size_t)r * NCOL + col] = acc[v] + bv;
    }
  }
}

// ---------------------------------------------------------------------------
// Elementwise / graph kernels
// ---------------------------------------------------------------------------
__global__ void zero_k(float* __restrict__ p, size_t n) {
  size_t i = (size_t)blockIdx.x * blockDim.x + threadIdx.x;
  size_t st = (size_t)gridDim.x * blockDim.x;
  for (; i < n; i += st) p[i] = 0.0f;
}

__global__ void bond_k(const float* __restrict__ r, float* __restrict__ d, int E) {
  int i = blockIdx.x * blockDim.x + threadIdx.x;
  if (i < E) {
    float a = r[3 * i], b = r[3 * i + 1], c = r[3 * i + 2];
    d[i] = sqrtf(a * a + b * b + c * c);
  }
}

// m[e,:] += p_src[src[e],:] + p_dst[dst[e],:]   (block per edge, thread per col)
__global__ void addgather_k(float* __restrict__ m, const float* __restrict__ ps,
                            const float* __restrict__ pd, const int* __restrict__ src,
                            const int* __restrict__ dst) {
  int e = blockIdx.x, c = threadIdx.x, C = blockDim.x;
  int s = src[e], d = dst[e];
  m[(size_t)e * C + c] += ps[(size_t)s * C + c] + pd[(size_t)d * C + c];
}

// sigma = sigmoid(m); num[dst] += du[src]*sigma; den[dst] += sigma
__global__ void scatter_k(const float* __restrict__ m, const float* __restrict__ du,
                          const int* __restrict__ src, const int* __restrict__ dst,
                          float* __restrict__ num, float* __restrict__ den) {
  int e = blockIdx.x, c = threadIdx.x, C = blockDim.x;
  int s = src[e], d = dst[e];
  float sg = sigmoid_f(m[(size_t)e * C + c]);
  atomicAdd(&num[(size_t)d * C + c], du[(size_t)s * C + c] * sg);
  atomicAdd(&den[(size_t)d * C + c], sg);
}

__global__ void combine_k(const float* __restrict__ su, const float* __restrict__ num,
                          const float* __restrict__ den, float* __restrict__ xout,
                          size_t total) {
  size_t i = (size_t)blockIdx.x * blockDim.x + threadIdx.x;
  size_t st = (size_t)gridDim.x * blockDim.x;
  for (; i < total; i += st) xout[i] = su[i] + num[i] / (den[i] + 1e-6f);
}

// column sums + sumsq (blockDim == C)
__global__ void colstats_k(const float* __restrict__ X, int M, int C,
                           float* __restrict__ acc) {
  int col = threadIdx.x;
  float s = 0.0f, sq = 0.0f;
  for (int r = blockIdx.x; r < M; r += gridDim.x) {
    float v = X[(size_t)r * C + col];
    s += v;
    sq += v * v;
  }
  atomicAdd(&acc[col], s);
  atomicAdd(&acc[C + col], sq);
}

// out = (residual? out:0) + act( g*(X-mu)*rsqrt(var+1e-5)+b ); biased var
__global__ void bnact_k(const float* __restrict__ X, const float* __restrict__ acc,
                        const float* __restrict__ g, const float* __restrict__ b,
                        float* __restrict__ out, int M, int C, int act, int residual) {
  size_t total = (size_t)M * C;
  size_t i = (size_t)blockIdx.x * blockDim.x + threadIdx.x;
  size_t st = (size_t)gridDim.x * blockDim.x;
  float invM = 1.0f / (float)M;
  for (; i < total; i += st) {
    int c = (int)(i % C);
    float mu = acc[c] * invM;
    float var = acc[C + c] * invM - mu * mu;
    float v = g[c] * (X[i] - mu) * rsqrtf(var + 1e-5f) + b[c];
    float a = (act == 1) ? softplus_f(v) : (act == 2) ? silu_f(v) : v;
    out[i] = residual ? (out[i] + a) : a;
  }
}

__global__ void poolscatter_k(const float* __restrict__ x, const int* __restrict__ gid,
                              float* __restrict__ pool, float* __restrict__ cnt) {
  int n = blockIdx.x, c = threadIdx.x, C = blockDim.x;
  int g = gid[n];
  atomicAdd(&pool[(size_t)g * C + c], x[(size_t)n * C + c]);
  if (c == 0) atomicAdd(&cnt[g], 1.0f);
}

__global__ void fc_k(const float* __restrict__ pool, const float* __restrict__ cnt,
                     const float* __restrict__ w, const float* __restrict__ b,
                     float* __restrict__ out, int G, int C) {
  int g = blockIdx.x * blockDim.x + threadIdx.x;
  if (g >= G) return;
  float inv = 1.0f / fmaxf(cnt[g], 1.0f);
  float s = b[0];
  for (int k = 0; k < C; k++) s += pool[(size_t)g * C + k] * inv * w[k];
  out[g] = s;
}

// ---------------------------------------------------------------------------
// Host orchestration
// ---------------------------------------------------------------------------
// Param leaves (insertion order of setup_inputs() pytree):
enum {
  P_ATOM = 0,      // lin.w, lin.b, bn.g, bn.b
  P_EDGE0 = 4, P_EDGE1 = 8,
  P_ANG0 = 12, P_ANG1 = 16,
  P_AL0_NODE = 20, P_AL0_EDGE = 34,
  P_AL1_NODE = 48, P_AL1_EDGE = 62,
  P_GCN0 = 76, P_GCN1 = 90,
  P_FC = 104,      // w[128], b[1]
  I_ATOMF = 106, I_R = 107, I_ANGLE = 108,
  I_SRC = 109, I_DST = 110, I_LSRC = 111, I_LDST = 112,
  I_GID = 113, I_NG = 114
};
// EGC leaf offsets: +0/1 src_gate w/b, +2/3 dst_gate, +4/5 edge_gate,
// +6/7 src_update, +8/9 dst_update, +10/11 bn_nodes g/b, +12/13 bn_edges g/b.

extern "C" void kernel_launch(void* const* d_in, const int* in_sizes, int n_in,
                              void* d_out, int out_size, void* d_ws, size_t ws_size,
                              hipStream_t stream) {
  (void)n_in; (void)ws_size;
  auto f = [&](int i) { return (const float*)d_in[i]; };
  auto ii = [&](int i) { return (const int*)d_in[i]; };

  const int N = in_sizes[I_ATOMF] / 92;
  const int E = in_sizes[I_R] / 3;
  const int LE = in_sizes[I_ANGLE];
  const int G = out_size;
  const int H = 128;

  float* ws = (float*)d_ws;
  size_t off = 0;
  auto alloc = [&](size_t n) { float* p = ws + off; off += n; return p; };
  float* x   = alloc((size_t)N * H);
  float* y   = alloc((size_t)E * H);
  float* z   = alloc((size_t)LE * H);
  float* m   = alloc((size_t)LE * H);   // messages; also angle-MLP intermediate
  float* p0  = alloc((size_t)E * H);    // src_gate proj; later xout
  float* p1  = alloc((size_t)E * H);    // dst_gate proj; later num
  float* p2  = alloc((size_t)E * H);    // dst_update proj
  float* p3  = alloc((size_t)E * H);    // src_update proj
  float* den = alloc((size_t)E * H);
  float* dlen = alloc((size_t)E);
  float* pool = alloc((size_t)G * H);
  float* cnt  = alloc((size_t)G);
  float* acc  = alloc(2 * (size_t)H);

  auto bnstage = [&](const float* X, int M, int C, const float* g_, const float* b_,
                     float* outp, int act, int residual) {
    zero_k<<<1, 256, 0, stream>>>(acc, (size_t)2 * C);
    int sblocks = M < 512 ? M : 512;
    colstats_k<<<sblocks, C, 0, stream>>>(X, M, C, acc);
    size_t tot = (size_t)M * C;
    int blocks = (int)((tot + 255) / 256);
    if (blocks > 4096) blocks = 4096;
    bnact_k<<<blocks, 256, 0, stream>>>(X, acc, g_, b_, outp, M, C, act, residual);
  };

  auto egc = [&](int P, const int* srcI, const int* dstI, float* h, float* e,
                 int Nn, int Ne) {
    int gh = (Nn + 63) / 64, ge = (Ne + 63) / 64;
    linear_wmma<128, 128, false><<<gh, 256, 0, stream>>>(h, f(P + 0), f(P + 1), p0, Nn, 0, 0, 0);
    linear_wmma<128, 128, false><<<gh, 256, 0, stream>>>(h, f(P + 2), f(P + 3), p1, Nn, 0, 0, 0);
    linear_wmma<128, 128, false><<<gh, 256, 0, stream>>>(h, f(P + 8), f(P + 9), p2, Nn, 0, 0, 0);
    linear_wmma<128, 128, false><<<gh, 256, 0, stream>>>(h, f(P + 6), f(P + 7), p3, Nn, 0, 0, 0);
    linear_wmma<128, 128, false><<<ge, 256, 0, stream>>>(e, f(P + 4), f(P + 5), m, Ne, 0, 0, 0);
    addgather_k<<<Ne, H, 0, stream>>>(m, p0, p1, srcI, dstI);
    zero_k<<<2048, 256, 0, stream>>>(p1, (size_t)Nn * H);   // num
    zero_k<<<2048, 256, 0, stream>>>(den, (size_t)Nn * H);
    scatter_k<<<Ne, H, 0, stream>>>(m, p2, srcI, dstI, p1, den);
    combine_k<<<4096, 256, 0, stream>>>(p3, p1, den, p0, (size_t)Nn * H);
    bnstage(p0, Nn, H, f(P + 10), f(P + 11), h, /*silu*/2, 1);
    bnstage(m, Ne, H, f(P + 12), f(P + 13), e, /*silu*/2, 1);
  };

  const float PI = 3.14159265358979323846f;

  // ---- atom embedding: x = softplus(bn(atom_features @ W + b))
  linear_wmma<92, 128, false><<<(N + 63) / 64, 256, 0, stream>>>(
      f(I_ATOMF), f(P_ATOM + 0), f(P_ATOM + 1), x, N, 0, 0, 0);
  bnstage(x, N, 128, f(P_ATOM + 2), f(P_ATOM + 3), x, 1, 0);

  // ---- edge embedding: rbf(|r|) -> mlp(80->64) -> mlp(64->128)
  bond_k<<<(E + 255) / 256, 256, 0, stream>>>(f(I_R), dlen, E);
  linear_wmma<80, 64, true><<<(E + 63) / 64, 256, 0, stream>>>(
      dlen, f(P_EDGE0 + 0), f(P_EDGE0 + 1), p0, E, 0.0f, 8.0f / 79.0f, 4.0f);
  bnstage(p0, E, 64, f(P_EDGE0 + 2), f(P_EDGE0 + 3), p0, 1, 0);
  linear_wmma<64, 128, false><<<(E + 63) / 64, 256, 0, stream>>>(
      p0, f(P_EDGE1 + 0), f(P_EDGE1 + 1), y, E, 0, 0, 0);
  bnstage(y, E, 128, f(P_EDGE1 + 2), f(P_EDGE1 + 3), y, 1, 0);

  // ---- angle embedding: rbf(angle) -> mlp(40->64) -> mlp(64->128)
  linear_wmma<40, 64, true><<<(LE + 63) / 64, 256, 0, stream>>>(
      f(I_ANGLE), f(P_ANG0 + 0), f(P_ANG0 + 1), m, LE, -PI / 2.0f, PI / 39.0f, 39.0f / PI);
  bnstage(m, LE, 64, f(P_ANG0 + 2), f(P_ANG0 + 3), m, 1, 0);
  linear_wmma<64, 128, false><<<(LE + 63) / 64, 256, 0, stream>>>(
      m, f(P_ANG1 + 0), f(P_ANG1 + 1), z, LE, 0, 0, 0);
  bnstage(z, LE, 128, f(P_ANG1 + 2), f(P_ANG1 + 3), z, 1, 0);

  // ---- ALIGNN layers (crystal-graph EGC, then line-graph EGC)
  egc(P_AL0_NODE, ii(I_SRC), ii(I_DST), x, y, N, E);
  egc(P_AL0_EDGE, ii(I_LSRC), ii(I_LDST), y, z, E, LE);
  egc(P_AL1_NODE, ii(I_SRC), ii(I_DST), x, y, N, E);
  egc(P_AL1_EDGE, ii(I_LSRC), ii(I_LDST), y, z, E, LE);

  // ---- GCN layers
  egc(P_GCN0, ii(I_SRC), ii(I_DST), x, y, N, E);
  egc(P_GCN1, ii(I_SRC), ii(I_DST), x, y, N, E);

  // ---- pooling + fc
  zero_k<<<64, 256, 0, stream>>>(pool, (size_t)G * H);
  zero_k<<<1, 64, 0, stream>>>(cnt, (size_t)G);
  poolscatter_k<<<N, H, 0, stream>>>(x, ii(I_GID), pool, cnt);
  fc_k<<<(G + 63) / 64, 64, 0, stream>>>(pool, cnt, f(P_FC), f(P_FC + 1),
                                         (float*)d_out, G, H);
}